// BrainAgeGAT_11287174053900
// MI455X (gfx1250) — compile-verified
//
#include <hip/hip_runtime.h>
#include <hip/hip_bf16.h>

#define N_NODES 50000
#define N_EDGES 800000
#define N_GRAPH 128
#define HC 256
#define NHEAD 8
#define CHEAD 32

// GEMM block tiling
#define MB 128          // rows per block (8 waves x 16)
#define NB 64           // cols per block (4 wmma subtiles)
#define KC 64           // K chunk staged in LDS
#define KP 68           // padded K stride (68 % 64 == 4 -> conflict-free frags)

typedef float v2f __attribute__((ext_vector_type(2)));
typedef float v8f __attribute__((ext_vector_type(8)));

__device__ __forceinline__ float lrelu(float x) { return x > 0.f ? x : 0.2f * x; }

__device__ __forceinline__ void atomicMaxF(float* addr, float v) {
    if (v >= 0.f) atomicMax((int*)addr, __float_as_int(v));
    else          atomicMin((unsigned int*)addr, (unsigned int)__float_as_int(v));
}

// ---------------- fill ----------------
__global__ void fill_f32(float* __restrict__ p, float v, long long n) {
    long long i = (long long)blockIdx.x * blockDim.x + threadIdx.x;
    if (i < n) p[i] = v;
}

// ---------------- layer-1 transform via WMMA (K=3 padded to 4) ----------------
// xl = x @ Wl1, xr = x @ Wr1 ; x:[N,3], W:[3,256]
__global__ void __launch_bounds__(256)
layer1_wmma(const float* __restrict__ x3, const float* __restrict__ Wl,
            const float* __restrict__ Wr, float* __restrict__ Cl,
            float* __restrict__ Cr, int M) {
    const int lane  = threadIdx.x & 31;
    const int wave  = threadIdx.x >> 5;
    const int tile  = blockIdx.x * 8 + wave;          // 16 rows x 64 cols per wave
    const int m0    = (tile >> 2) * 16;               // 256/64 = 4 col-tiles
    const int n0    = (tile & 3) * 64;
    if (m0 >= M) return;
    const int row   = lane & 15;
    const int khalf = lane >> 4;                      // 0: K={0,1}, 1: K={2,3}

    // A fragment: 16x4 f32, K=3 zero-padded. Loads hoisted (all in-bounds) so the
    // khalf select lowers to v_cndmask, not exec-masked branches.
    const float ax = x3[(size_t)(m0 + row) * 3 + 2 * khalf];  // K=0 or K=2
    const float ay = x3[(size_t)(m0 + row) * 3 + 1];          // K=1 (safe for all rows)
    v2f a;
    a.x = ax;
    a.y = khalf ? 0.f : ay;

    v8f accl[4], accr[4];
#pragma unroll
    for (int t = 0; t < 4; ++t) {
        const int col = n0 + t * 16 + row;
        const float blx = Wl[(2 * khalf) * HC + col];
        const float bly = Wl[1 * HC + col];
        const float brx = Wr[(2 * khalf) * HC + col];
        const float bry = Wr[1 * HC + col];
        v2f bl, br;
        bl.x = blx; bl.y = khalf ? 0.f : bly;
        br.x = brx; br.y = khalf ? 0.f : bry;
        v8f z = {};
        accl[t] = __builtin_amdgcn_wmma_f32_16x16x4_f32(false, a, false, bl, (short)0, z, false, false);
        accr[t] = __builtin_amdgcn_wmma_f32_16x16x4_f32(false, a, false, br, (short)0, z, false, false);
    }
#pragma unroll
    for (int t = 0; t < 4; ++t)
#pragma unroll
        for (int v = 0; v < 8; ++v) {
            const size_t o = (size_t)(m0 + v + 8 * khalf) * HC + n0 + t * 16 + row;
            Cl[o] = accl[t][v];
            Cr[o] = accr[t][v];
        }
}

// ---------------- layer-2 dual GEMM via WMMA f32, LDS-staged B panels ----------
// Cl = A @ Bl, Cr = A @ Br ; A:[M,256], B:[256,256] (exact f32 path)
// Block: 128x64 tile, 8 waves of 16 rows each; B panels staged K-major in LDS.
__global__ void __launch_bounds__(256)
gemm_dual_wmma(const float* __restrict__ A, const float* __restrict__ Bl,
               const float* __restrict__ Br, float* __restrict__ Cl,
               float* __restrict__ Cr, int M) {
    __shared__ float sB[2][NB][KP];                   // 2 * 64 * 68 * 4 = 34 KB

    const int n0 = blockIdx.y * NB;
    int m0 = blockIdx.x * MB + (threadIdx.x >> 5) * 16;
    if (m0 + 16 > M) m0 = M - 16;                     // tail: duplicate tile, same values
    const int lane  = threadIdx.x & 31;
    const int row   = lane & 15;
    const int khalf = lane >> 4;

    const int sc = threadIdx.x & 63;                  // staging col
    const int sk = (threadIdx.x >> 6) * 16;           // staging k-group base

    v8f accl[4] = {}, accr[4] = {};
    const float* arow = A + (size_t)(m0 + row) * HC + 2 * khalf;

    for (int kc = 0; kc < HC; kc += KC) {
        // stage B[kc..kc+63, n0..n0+63] for both matrices, K-major
#pragma unroll 4
        for (int kk = 0; kk < 16; ++kk) {
            const int k = sk + kk;
            sB[0][sc][k] = Bl[(size_t)(kc + k) * HC + n0 + sc];
            sB[1][sc][k] = Br[(size_t)(kc + k) * HC + n0 + sc];
        }
        __syncthreads();

        for (int k0 = 0; k0 < KC; k0 += 4) {
            v2f a = *(const v2f*)(arow + kc + k0);    // A frag: K=k0+2*khalf, +1
            const int krow = k0 + 2 * khalf;
#pragma unroll
            for (int t = 0; t < 4; ++t) {
                const int col = t * 16 + row;
                v2f bl = *(const v2f*)&sB[0][col][krow];
                v2f br = *(const v2f*)&sB[1][col][krow];
                accl[t] = __builtin_amdgcn_wmma_f32_16x16x4_f32(false, a, false, bl, (short)0, accl[t], false, false);
                accr[t] = __builtin_amdgcn_wmma_f32_16x16x4_f32(false, a, false, br, (short)0, accr[t], false, false);
            }
        }
        __syncthreads();
    }

#pragma unroll
    for (int t = 0; t < 4; ++t)
#pragma unroll
        for (int v = 0; v < 8; ++v) {
            const size_t o = (size_t)(m0 + v + 8 * khalf) * HC + n0 + t * 16 + row;
            Cl[o] = accl[t][v];
            Cr[o] = accr[t][v];
        }
}

// ---------------- edge pass A: logits + segment max ----------------
__global__ void __launch_bounds__(256)
edge_logits_max(const float* __restrict__ xl, const float* __restrict__ xr,
                const int* __restrict__ esrc, const int* __restrict__ edst,
                const float* __restrict__ att, float* __restrict__ logits,
                float* __restrict__ mmax, int Etot, int Eorig) {
    __shared__ float satt[HC];
    satt[threadIdx.x] = att[threadIdx.x];
    __syncthreads();
    const int idx = blockIdx.x * 256 + threadIdx.x;   // (edge, head)
    const int e = idx >> 3, h = idx & 7;
    if (e >= Etot) return;
    const int s = (e < Eorig) ? esrc[e] : (e - Eorig);
    const int d = (e < Eorig) ? edst[e] : (e - Eorig);
    const float4* pl = (const float4*)(xl + (size_t)s * HC + h * CHEAD);
    const float4* pr = (const float4*)(xr + (size_t)d * HC + h * CHEAD);
    const float4* pa = (const float4*)(satt + h * CHEAD);
    float acc = 0.f;
#pragma unroll
    for (int i = 0; i < 8; ++i) {
        float4 a = pl[i], b = pr[i], w = pa[i];
        acc += w.x * lrelu(a.x + b.x);
        acc += w.y * lrelu(a.y + b.y);
        acc += w.z * lrelu(a.z + b.z);
        acc += w.w * lrelu(a.w + b.w);
    }
    logits[idx] = acc;
    atomicMaxF(&mmax[d * NHEAD + h], acc);
}

// ---------------- edge pass B: exp + segment sum ----------------
__global__ void __launch_bounds__(256)
edge_exp_denom(const int* __restrict__ edst, const float* __restrict__ mmax,
               float* __restrict__ logits, float* __restrict__ denom,
               int Etot, int Eorig) {
    const int idx = blockIdx.x * 256 + threadIdx.x;
    const int e = idx >> 3, h = idx & 7;
    if (e >= Etot) return;
    const int d = (e < Eorig) ? edst[e] : (e - Eorig);
    const float ex = __expf(logits[idx] - mmax[d * NHEAD + h]);
    logits[idx] = ex;                                  // reuse buffer for exp values
    atomicAdd(&denom[d * NHEAD + h], ex);
}

// ---------------- edge pass C: alpha-weighted aggregation ----------------
__global__ void __launch_bounds__(256)
edge_aggregate(const float* __restrict__ xl, const int* __restrict__ esrc,
               const int* __restrict__ edst, const float* __restrict__ ex,
               const float* __restrict__ denom, float* __restrict__ out,
               int Etot, int Eorig) {
    const int idx = blockIdx.x * 256 + threadIdx.x;
    const int e = idx >> 3, h = idx & 7;
    if (e >= Etot) return;
    const int s = (e < Eorig) ? esrc[e] : (e - Eorig);
    const int d = (e < Eorig) ? edst[e] : (e - Eorig);
    const float alpha = ex[idx] / denom[d * NHEAD + h];
    const float* sp = xl + (size_t)s * HC + h * CHEAD;
    float* dp = out + (size_t)d * HC + h * CHEAD;
#pragma unroll
    for (int c = 0; c < CHEAD; ++c)
        atomicAdd(&dp[c], alpha * sp[c]);
}

// ---------------- bias + relu (in place) ----------------
__global__ void finish_relu_bias(float* __restrict__ h, const float* __restrict__ b, long long n) {
    long long i = (long long)blockIdx.x * blockDim.x + threadIdx.x;
    if (i >= n) return;
    float v = h[i] + b[i & (HC - 1)];
    h[i] = v > 0.f ? v : 0.f;
}

// ---------------- global mean pool (accumulate) ----------------
__global__ void __launch_bounds__(256)
pool_add(const float* __restrict__ h, const int* __restrict__ batch,
         float* __restrict__ pooled, float* __restrict__ counts, int n) {
    const int node = blockIdx.x;           // blockDim == 256 == HC
    if (node >= n) return;
    const int g = batch[node];
    const int c = threadIdx.x;
    atomicAdd(&pooled[(size_t)g * HC + c], h[(size_t)node * HC + c]);
    if (c == 0) atomicAdd(&counts[g], 1.0f);
}

// ---------------- MLP head ----------------
__global__ void __launch_bounds__(64)
mlp_head(const float* __restrict__ pooled, const float* __restrict__ counts,
         const float* __restrict__ u, const float* __restrict__ W1,
         const float* __restrict__ b1, const float* __restrict__ W2,
         const float* __restrict__ b2, float* __restrict__ out) {
    const int g = blockIdx.x, j = threadIdx.x;
    __shared__ float a[64];
    float cnt = counts[g];
    const float inv = 1.f / (cnt > 1.f ? cnt : 1.f);
    float s = b1[j];
    const float* pg = pooled + (size_t)g * HC;
    for (int k = 0; k < HC; ++k) s += (pg[k] * inv) * W1[k * 64 + j];
    a[j] = s > 0.f ? s : 0.f;
    __syncthreads();
    if (j == 0) {
        float o = b2[0];
        for (int t = 0; t < 64; ++t) o += a[t] * W2[t];
        for (int t = 0; t < 3; ++t)  o += u[g * 3 + t] * W2[64 + t];
        out[g] = o;
    }
}

extern "C" void kernel_launch(void* const* d_in, const int* in_sizes, int n_in,
                              void* d_out, int out_size, void* d_ws, size_t ws_size,
                              hipStream_t stream) {
    const float* x     = (const float*)d_in[0];
    const int*   esrc  = (const int*)d_in[1];
    const int*   edst  = esrc + N_EDGES;
    const int*   batch = (const int*)d_in[2];
    const float* u     = (const float*)d_in[3];
    const float* Wl1   = (const float*)d_in[4];
    const float* Wr1   = (const float*)d_in[5];
    const float* att1  = (const float*)d_in[6];
    const float* b1    = (const float*)d_in[7];
    const float* Wl2   = (const float*)d_in[8];
    const float* Wr2   = (const float*)d_in[9];
    const float* att2  = (const float*)d_in[10];
    const float* b2    = (const float*)d_in[11];
    const float* Wlin1 = (const float*)d_in[12];
    const float* blin1 = (const float*)d_in[13];
    const float* Wout  = (const float*)d_in[14];
    const float* bout  = (const float*)d_in[15];
    float* out = (float*)d_out;

    const int Etot = N_EDGES + N_NODES;               // with self loops
    const long long NF = (long long)N_NODES * HC;     // 12.8M

    // workspace carve-up (all 256B-aligned sizes)
    char* w = (char*)d_ws;
    float* XL = (float*)w;               w += (size_t)NF * 4;
    float* XR = (float*)w;               w += (size_t)NF * 4;
    float* AG = (float*)w;               w += (size_t)NF * 4;          // agg / h
    float* LG = (float*)w;               w += (size_t)Etot * NHEAD * 4; // logits->exp
    float* MX = (float*)w;               w += (size_t)N_NODES * NHEAD * 4;
    float* DN = (float*)w;               w += (size_t)N_NODES * NHEAD * 4;
    float* PL = (float*)w;               w += (size_t)N_GRAPH * HC * 4;
    float* CT = (float*)w;               w += 512;

    const int gNF   = (int)((NF + 255) / 256);
    const int gNH   = (N_NODES * NHEAD + 255) / 256;
    const int gEH   = (Etot * NHEAD + 255) / 256;
    const int gT1   = (N_NODES / 16 * 4 + 7) / 8;     // layer-1 wave-tiles / 8 waves
    const dim3 gGemm((N_NODES + MB - 1) / MB, HC / NB);
    const float NEG_INF = -__builtin_huge_valf();

    // ---- init ----
    fill_f32<<<gNF, 256, 0, stream>>>(AG, 0.f, NF);
    fill_f32<<<gNH, 256, 0, stream>>>(MX, NEG_INF, (long long)N_NODES * NHEAD);
    fill_f32<<<gNH, 256, 0, stream>>>(DN, 0.f, (long long)N_NODES * NHEAD);
    fill_f32<<<(N_GRAPH * HC + 255) / 256, 256, 0, stream>>>(PL, 0.f, N_GRAPH * HC);
    fill_f32<<<1, 256, 0, stream>>>(CT, 0.f, N_GRAPH);

    // ---- layer 1 ----
    layer1_wmma<<<gT1, 256, 0, stream>>>(x, Wl1, Wr1, XL, XR, N_NODES);
    edge_logits_max<<<gEH, 256, 0, stream>>>(XL, XR, esrc, edst, att1, LG, MX, Etot, N_EDGES);
    edge_exp_denom<<<gEH, 256, 0, stream>>>(edst, MX, LG, DN, Etot, N_EDGES);
    edge_aggregate<<<gEH, 256, 0, stream>>>(XL, esrc, edst, LG, DN, AG, Etot, N_EDGES);
    finish_relu_bias<<<gNF, 256, 0, stream>>>(AG, b1, NF);   // AG = h1

    // ---- layer 2 ----
    fill_f32<<<gNH, 256, 0, stream>>>(MX, NEG_INF, (long long)N_NODES * NHEAD);
    fill_f32<<<gNH, 256, 0, stream>>>(DN, 0.f, (long long)N_NODES * NHEAD);
    gemm_dual_wmma<<<gGemm, 256, 0, stream>>>(AG, Wl2, Wr2, XL, XR, N_NODES);
    fill_f32<<<gNF, 256, 0, stream>>>(AG, 0.f, NF);          // reuse for layer-2 agg
    edge_logits_max<<<gEH, 256, 0, stream>>>(XL, XR, esrc, edst, att2, LG, MX, Etot, N_EDGES);
    edge_exp_denom<<<gEH, 256, 0, stream>>>(edst, MX, LG, DN, Etot, N_EDGES);
    edge_aggregate<<<gEH, 256, 0, stream>>>(XL, esrc, edst, LG, DN, AG, Etot, N_EDGES);
    finish_relu_bias<<<gNF, 256, 0, stream>>>(AG, b2, NF);   // AG = h2

    // ---- pool + head ----
    pool_add<<<N_NODES, 256, 0, stream>>>(AG, batch, PL, CT, N_NODES);
    mlp_head<<<N_GRAPH, 64, 0, stream>>>(PL, CT, u, Wlin1, blin1, Wout, bout, out);
}